// CodebookGNN_66563403153851
// MI455X (gfx1250) — compile-verified
//
#include <hip/hip_runtime.h>
#include <math.h>

typedef __attribute__((ext_vector_type(2))) float v2f;
typedef __attribute__((ext_vector_type(8))) float v8f;

#define HIDC 256
#define DOUTC 32

// ---------------------------------------------------------------------------
// zero-fill (graph-capture-safe init of accumulators)
// ---------------------------------------------------------------------------
__global__ __launch_bounds__(256) void k_zero(float* __restrict__ p, int n) {
    int i = blockIdx.x * 256 + threadIdx.x;
    if (i < n) p[i] = 0.f;
}

// ---------------------------------------------------------------------------
// deterministic mean of edge_attr (two-stage tree reduction)
// ---------------------------------------------------------------------------
__global__ __launch_bounds__(256) void k_sum_partial(const float* __restrict__ x, int n,
                                                     float* __restrict__ partial) {
    __shared__ float sm[256];
    int tid = threadIdx.x;
    float s = 0.f;
    for (int i = blockIdx.x * 256 + tid; i < n; i += gridDim.x * 256) s += x[i];
    sm[tid] = s;
    __syncthreads();
    for (int o = 128; o > 0; o >>= 1) {
        if (tid < o) sm[tid] += sm[tid + o];
        __syncthreads();
    }
    if (tid == 0) partial[blockIdx.x] = sm[0];
}

__global__ __launch_bounds__(256) void k_sum_final(const float* __restrict__ partial,
                                                   float* __restrict__ out, float inv_n) {
    __shared__ float sm[256];
    int tid = threadIdx.x;
    sm[tid] = partial[tid];
    __syncthreads();
    for (int o = 128; o > 0; o >>= 1) {
        if (tid < o) sm[tid] += sm[tid + o];
        __syncthreads();
    }
    if (tid == 0) out[0] = sm[0] * inv_n;
}

// ---------------------------------------------------------------------------
// fp32 WMMA GEMM: C[M,N] = A[M,K] @ B[K,N] + bias[N], N/K compile-time.
// One wave computes a 32Mx32N block = 2x2 WMMA tiles (V_WMMA_F32_16X16X4_F32).
// A-frag: lane L (M = L&15) holds {A[m][k0+kh], A[m][k0+kh+1]}, kh = (L>>4)*2
// B-frag: components hold rows K = k0+kh(+1), col = tile*16 + (L&15)
// C-frag: acc[v]  = C[tileM*16 + v + (L>>4)*8][tileN*16 + (L&15)]
// M-tail handled branchlessly: clamp row pointer, multiply A frag by 0/1 mask.
// ---------------------------------------------------------------------------
template <int N, int K>
__global__ __launch_bounds__(32) void k_gemm_f32_wmma(const float* __restrict__ A,
                                                      const float* __restrict__ B,
                                                      const float* __restrict__ bias,
                                                      float* __restrict__ C, int M) {
    const int lane  = threadIdx.x;       // 0..31
    const int l15   = lane & 15;
    const int khalf = (lane >> 4) * 2;   // 0 or 2
    const int tm    = blockIdx.x;        // 32-row block
    const int tn    = blockIdx.y;        // 32-col block

    const int m0 = tm * 32 + l15;
    const int m1 = m0 + 16;
    const int n0 = tn * 32 + l15;
    const int n1 = n0 + 16;

    const float msk0 = (m0 < M) ? 1.f : 0.f;
    const float msk1 = (m1 < M) ? 1.f : 0.f;
    const float* ar0 = A + (size_t)((m0 < M) ? m0 : (M - 1)) * K + khalf;
    const float* ar1 = A + (size_t)((m1 < M) ? m1 : (M - 1)) * K + khalf;

    v8f acc00 = {}, acc01 = {}, acc10 = {}, acc11 = {};

#pragma unroll 4
    for (int k0 = 0; k0 < K; k0 += 4) {
        float2 a0f = *(const float2*)(ar0 + k0);
        float2 a1f = *(const float2*)(ar1 + k0);
        v2f a0, a1, b0, b1;
        a0.x = a0f.x * msk0; a0.y = a0f.y * msk0;
        a1.x = a1f.x * msk1; a1.y = a1f.y * msk1;
        const float* bpk = B + (size_t)(k0 + khalf) * N;  // N is constant: shifts only
        b0.x = bpk[n0];
        b0.y = bpk[n0 + N];
        b1.x = bpk[n1];
        b1.y = bpk[n1 + N];
        acc00 = __builtin_amdgcn_wmma_f32_16x16x4_f32(false, a0, false, b0, (short)0, acc00, false, false);
        acc01 = __builtin_amdgcn_wmma_f32_16x16x4_f32(false, a0, false, b1, (short)0, acc01, false, false);
        acc10 = __builtin_amdgcn_wmma_f32_16x16x4_f32(false, a1, false, b0, (short)0, acc10, false, false);
        acc11 = __builtin_amdgcn_wmma_f32_16x16x4_f32(false, a1, false, b1, (short)0, acc11, false, false);
    }

    const float bv0 = bias[n0];
    const float bv1 = bias[n1];
    const int mb0 = tm * 32 + (lane >> 4) * 8;  // rows for acc0x
    const int mb1 = mb0 + 16;                   // rows for acc1x
#pragma unroll
    for (int v = 0; v < 8; ++v) {
        int ma = mb0 + v;
        if (ma < M) {
            C[(size_t)ma * N + n0] = acc00[v] + bv0;
            C[(size_t)ma * N + n1] = acc01[v] + bv1;
        }
        int mb = mb1 + v;
        if (mb < M) {
            C[(size_t)mb * N + n0] = acc10[v] + bv0;
            C[(size_t)mb * N + n1] = acc11[v] + bv1;
        }
    }
}

// ---------------------------------------------------------------------------
// edge helpers
// ---------------------------------------------------------------------------
__device__ __forceinline__ void edge_sd(int e, int E, const int* __restrict__ srcI,
                                        const int* __restrict__ dstI, int& s, int& d) {
    if (e < E) { s = srcI[e]; d = dstI[e]; }
    else       { s = d = e - E; }
}

__device__ __forceinline__ unsigned f32_enc(float f) {
    unsigned u = __float_as_uint(f);
    return (u & 0x80000000u) ? ~u : (u | 0x80000000u);
}
__device__ __forceinline__ float f32_dec(unsigned e) {
    unsigned u = (e & 0x80000000u) ? (e & 0x7fffffffu) : ~e;
    return __uint_as_float(u);
}

// ---------------------------------------------------------------------------
// GATv2 attention logits, 256-channel layer (wave/edge, 8 ch/lane, 2 heads)
// logit[e][h] = sum_c att[h][c] * lrelu(xr[dst]+xl[src]+ea*We, 0.2)
// ---------------------------------------------------------------------------
__global__ __launch_bounds__(256) void k_edge_logits256(
    const float* __restrict__ xl, const float* __restrict__ xr,
    const int* __restrict__ srcI, const int* __restrict__ dstI,
    const float* __restrict__ eattr, const float* __restrict__ We,
    const float* __restrict__ att, const float* __restrict__ ea_mean,
    float* __restrict__ logits, int E, int Nn) {
    const int wid  = blockIdx.x * 8 + (threadIdx.x >> 5);
    const int lane = threadIdx.x & 31;
    if (wid >= E + Nn) return;
    int s, d;
    edge_sd(wid, E, srcI, dstI, s, d);
    const float ea = (wid < E) ? eattr[wid] : ea_mean[0];

    const float4* xl4 = (const float4*)(xl + (size_t)s * HIDC) + lane * 2;
    const float4* xr4 = (const float4*)(xr + (size_t)d * HIDC) + lane * 2;
    const float4* we4 = (const float4*)We + lane * 2;
    const float4* at4 = (const float4*)att + lane * 2;

    float acc = 0.f;
#pragma unroll
    for (int p = 0; p < 2; ++p) {
        float4 a = xl4[p], b = xr4[p], w = we4[p], t = at4[p];
        float v;
        v = b.x + a.x + ea * w.x; v = v > 0.f ? v : 0.2f * v; acc += t.x * v;
        v = b.y + a.y + ea * w.y; v = v > 0.f ? v : 0.2f * v; acc += t.y * v;
        v = b.z + a.z + ea * w.z; v = v > 0.f ? v : 0.2f * v; acc += t.z * v;
        v = b.w + a.w + ea * w.w; v = v > 0.f ? v : 0.2f * v; acc += t.w * v;
    }
#pragma unroll
    for (int o = 1; o < 16; o <<= 1) acc += __shfl_xor(acc, o, 32);
    if ((lane & 15) == 0) logits[(size_t)wid * 2 + (lane >> 4)] = acc;
}

// 32-channel layer (wave/edge, 1 ch/lane)
__global__ __launch_bounds__(256) void k_edge_logits32(
    const float* __restrict__ xl, const float* __restrict__ xr,
    const int* __restrict__ srcI, const int* __restrict__ dstI,
    const float* __restrict__ eattr, const float* __restrict__ We,
    const float* __restrict__ att, const float* __restrict__ ea_mean,
    float* __restrict__ logits, int E, int Nn) {
    const int wid  = blockIdx.x * 8 + (threadIdx.x >> 5);
    const int lane = threadIdx.x & 31;
    if (wid >= E + Nn) return;
    int s, d;
    edge_sd(wid, E, srcI, dstI, s, d);
    const float ea = (wid < E) ? eattr[wid] : ea_mean[0];
    float v = xr[(size_t)d * DOUTC + lane] + xl[(size_t)s * DOUTC + lane] + ea * We[lane];
    v = v > 0.f ? v : 0.2f * v;
    float acc = att[lane] * v;
#pragma unroll
    for (int o = 1; o < 16; o <<= 1) acc += __shfl_xor(acc, o, 32);
    if ((lane & 15) == 0) logits[(size_t)wid * 2 + (lane >> 4)] = acc;
}

// ---------------------------------------------------------------------------
// segment softmax: max pass, then exp/sum pass (thread per edge*head)
// ---------------------------------------------------------------------------
__global__ __launch_bounds__(256) void k_seg_max(const float* __restrict__ logits,
                                                 const int* __restrict__ dstI,
                                                 unsigned* __restrict__ m, int E, int Nn) {
    int i = blockIdx.x * 256 + threadIdx.x;
    if (i >= (E + Nn) * 2) return;
    int e = i >> 1, h = i & 1;
    int d = (e < E) ? dstI[e] : (e - E);
    atomicMax(&m[d * 2 + h], f32_enc(logits[i]));
}

__global__ __launch_bounds__(256) void k_seg_expsum(float* __restrict__ logits,
                                                    const int* __restrict__ dstI,
                                                    const unsigned* __restrict__ m,
                                                    float* __restrict__ ssum, int E, int Nn) {
    int i = blockIdx.x * 256 + threadIdx.x;
    if (i >= (E + Nn) * 2) return;
    int e = i >> 1, h = i & 1;
    int d = (e < E) ? dstI[e] : (e - E);
    float ex = expf(logits[i] - f32_dec(m[d * 2 + h]));
    logits[i] = ex;  // overwrite logit with exp
    atomicAdd(&ssum[d * 2 + h], ex);
}

// ---------------------------------------------------------------------------
// weighted scatter aggregation: out[dst] += alpha * xl[src]
// ---------------------------------------------------------------------------
__global__ __launch_bounds__(256) void k_agg256(const float* __restrict__ xl,
                                                const int* __restrict__ srcI,
                                                const int* __restrict__ dstI,
                                                const float* __restrict__ ex,
                                                const float* __restrict__ ssum,
                                                float* __restrict__ agg, int E, int Nn) {
    const int wid  = blockIdx.x * 8 + (threadIdx.x >> 5);
    const int lane = threadIdx.x & 31;
    if (wid >= E + Nn) return;
    int s, d;
    edge_sd(wid, E, srcI, dstI, s, d);
    const int h = lane >> 4;
    const float w = ex[(size_t)wid * 2 + h] / (ssum[d * 2 + h] + 1e-16f);
    const float4* xs = (const float4*)(xl + (size_t)s * HIDC) + lane * 2;
    float4 a = xs[0], b = xs[1];
    float* od = agg + (size_t)d * HIDC + lane * 8;
    atomicAdd(od + 0, w * a.x); atomicAdd(od + 1, w * a.y);
    atomicAdd(od + 2, w * a.z); atomicAdd(od + 3, w * a.w);
    atomicAdd(od + 4, w * b.x); atomicAdd(od + 5, w * b.y);
    atomicAdd(od + 6, w * b.z); atomicAdd(od + 7, w * b.w);
}

__global__ __launch_bounds__(256) void k_agg32(const float* __restrict__ xl,
                                               const int* __restrict__ srcI,
                                               const int* __restrict__ dstI,
                                               const float* __restrict__ ex,
                                               const float* __restrict__ ssum,
                                               float* __restrict__ agg, int E, int Nn) {
    const int wid  = blockIdx.x * 8 + (threadIdx.x >> 5);
    const int lane = threadIdx.x & 31;
    if (wid >= E + Nn) return;
    int s, d;
    edge_sd(wid, E, srcI, dstI, s, d);
    const int h = lane >> 4;
    const float w = ex[(size_t)wid * 2 + h] / (ssum[d * 2 + h] + 1e-16f);
    atomicAdd(&agg[(size_t)d * DOUTC + lane], w * xl[(size_t)s * DOUTC + lane]);
}

// ---------------------------------------------------------------------------
// post layer0: h = relu(LN(agg + bias + prev))    (wave/node, 8 ch/lane)
// ---------------------------------------------------------------------------
__global__ __launch_bounds__(256) void k_post0(const float* __restrict__ agg,
                                               const float* __restrict__ bias,
                                               const float* __restrict__ prev,
                                               const float* __restrict__ g,
                                               const float* __restrict__ be,
                                               float* __restrict__ out, int Nn) {
    const int n    = blockIdx.x * 8 + (threadIdx.x >> 5);
    const int lane = threadIdx.x & 31;
    if (n >= Nn) return;
    const float* ap = agg  + (size_t)n * HIDC + lane * 8;
    const float* pp = prev + (size_t)n * HIDC + lane * 8;
    const float* bb = bias + lane * 8;
    float v[8], s = 0.f, s2 = 0.f;
#pragma unroll
    for (int j = 0; j < 8; ++j) { v[j] = ap[j] + bb[j] + pp[j]; s += v[j]; s2 += v[j] * v[j]; }
#pragma unroll
    for (int o = 1; o < 32; o <<= 1) { s += __shfl_xor(s, o, 32); s2 += __shfl_xor(s2, o, 32); }
    const float mu  = s * (1.f / 256.f);
    const float var = s2 * (1.f / 256.f) - mu * mu;
    const float r   = rsqrtf(var + 1e-5f);
    const float* gg = g + lane * 8;
    const float* eb = be + lane * 8;
    float* op = out + (size_t)n * HIDC + lane * 8;
#pragma unroll
    for (int j = 0; j < 8; ++j) {
        float y = (v[j] - mu) * r * gg[j] + eb[j];
        op[j] = fmaxf(y, 0.f);
    }
}

// final: out = relu(LN(agg + bias))   (wave/node, 1 ch/lane over 32 ch)
__global__ __launch_bounds__(256) void k_final(const float* __restrict__ agg,
                                               const float* __restrict__ bias,
                                               const float* __restrict__ g,
                                               const float* __restrict__ be,
                                               float* __restrict__ out, int Nn) {
    const int n    = blockIdx.x * 8 + (threadIdx.x >> 5);
    const int lane = threadIdx.x & 31;
    if (n >= Nn) return;
    float v = agg[(size_t)n * DOUTC + lane] + bias[lane];
    float s = v, s2 = v * v;
#pragma unroll
    for (int o = 1; o < 32; o <<= 1) { s += __shfl_xor(s, o, 32); s2 += __shfl_xor(s2, o, 32); }
    const float mu  = s * (1.f / 32.f);
    const float var = s2 * (1.f / 32.f) - mu * mu;
    const float r   = rsqrtf(var + 1e-5f);
    float y = (v - mu) * r * g[lane] + be[lane];
    out[(size_t)n * DOUTC + lane] = fmaxf(y, 0.f);
}

// ---------------------------------------------------------------------------
extern "C" void kernel_launch(void* const* d_in, const int* in_sizes, int n_in,
                              void* d_out, int out_size, void* d_ws, size_t ws_size,
                              hipStream_t stream) {
    const float* x     = (const float*)d_in[0];
    const int*   ei    = (const int*)  d_in[1];
    const float* eattr = (const float*)d_in[2];
    const float* Wp    = (const float*)d_in[3];
    const float* bp    = (const float*)d_in[4];
    const float* Wl0   = (const float*)d_in[5];
    const float* bl0   = (const float*)d_in[6];
    const float* Wr0   = (const float*)d_in[7];
    const float* br0   = (const float*)d_in[8];
    const float* We0   = (const float*)d_in[9];
    const float* att0  = (const float*)d_in[10];
    const float* bias0 = (const float*)d_in[11];
    const float* g0    = (const float*)d_in[12];
    const float* be0   = (const float*)d_in[13];
    const float* Wl1   = (const float*)d_in[14];
    const float* bl1   = (const float*)d_in[15];
    const float* Wr1   = (const float*)d_in[16];
    const float* br1   = (const float*)d_in[17];
    const float* We1   = (const float*)d_in[18];
    const float* att1  = (const float*)d_in[19];
    const float* bias1 = (const float*)d_in[20];
    const float* g1    = (const float*)d_in[21];
    const float* be1   = (const float*)d_in[22];

    const int N    = in_sizes[0] / 512;  // 10000
    const int E    = in_sizes[1] / 2;    // 320000
    const int Etot = E + N;
    const int* srcI = ei;
    const int* dstI = ei + E;

    // workspace carve-out (256B aligned regions)
    char* wp = (char*)d_ws;
    auto carve = [&](size_t bytes) -> void* {
        void* p = (void*)wp;
        wp += (bytes + 255) & ~(size_t)255;
        return p;
    };
    float*    hidden  = (float*)   carve((size_t)N * HIDC * 4);
    float*    xl0     = (float*)   carve((size_t)N * HIDC * 4);
    float*    xr0     = (float*)   carve((size_t)N * HIDC * 4);
    float*    agg0    = (float*)   carve((size_t)N * HIDC * 4);
    float*    h0      = (float*)   carve((size_t)N * HIDC * 4);
    float*    xl1     = (float*)   carve((size_t)N * DOUTC * 4);
    float*    xr1     = (float*)   carve((size_t)N * DOUTC * 4);
    float*    agg1    = (float*)   carve((size_t)N * DOUTC * 4);
    float*    logits  = (float*)   carve((size_t)Etot * 2 * 4);
    unsigned* segm    = (unsigned*)carve((size_t)N * 2 * 4);
    float*    segs    = (float*)   carve((size_t)N * 2 * 4);
    float*    partial = (float*)   carve(256 * 4);
    float*    ea_mean = (float*)   carve(4);

    const int TPB   = 256;
    const int gridE = (Etot * 2 + TPB - 1) / TPB;   // thread per (edge,head)
    const int gridW = (Etot + 7) / 8;               // wave per edge
    const int gridN = (N + 7) / 8;                  // wave per node
    const int gridM = (N + 31) / 32;                // 32-row GEMM blocks

    // edge_attr mean (deterministic tree reduction)
    k_sum_partial<<<256, 256, 0, stream>>>(eattr, E, partial);
    k_sum_final<<<1, 256, 0, stream>>>(partial, ea_mean, 1.f / (float)E);

    // hidden = x @ Wp + bp
    k_gemm_f32_wmma<HIDC, 512><<<dim3(gridM, HIDC / 32), 32, 0, stream>>>(x, Wp, bp, hidden, N);
    // xl0 = hidden @ Wl0 + bl0 ; xr0 = hidden @ Wr0 + br0
    k_gemm_f32_wmma<HIDC, HIDC><<<dim3(gridM, HIDC / 32), 32, 0, stream>>>(hidden, Wl0, bl0, xl0, N);
    k_gemm_f32_wmma<HIDC, HIDC><<<dim3(gridM, HIDC / 32), 32, 0, stream>>>(hidden, Wr0, br0, xr0, N);

    // ---- layer 0 attention ----
    k_zero<<<((N * HIDC) + TPB - 1) / TPB, TPB, 0, stream>>>(agg0, N * HIDC);
    k_zero<<<((N * 2) + TPB - 1) / TPB, TPB, 0, stream>>>((float*)segm, N * 2);
    k_zero<<<((N * 2) + TPB - 1) / TPB, TPB, 0, stream>>>(segs, N * 2);

    k_edge_logits256<<<gridW, TPB, 0, stream>>>(xl0, xr0, srcI, dstI, eattr, We0, att0,
                                                ea_mean, logits, E, N);
    k_seg_max<<<gridE, TPB, 0, stream>>>(logits, dstI, segm, E, N);
    k_seg_expsum<<<gridE, TPB, 0, stream>>>(logits, dstI, segm, segs, E, N);
    k_agg256<<<gridW, TPB, 0, stream>>>(xl0, srcI, dstI, logits, segs, agg0, E, N);

    // h0 = relu(LN(agg0 + bias0 + hidden))
    k_post0<<<gridN, TPB, 0, stream>>>(agg0, bias0, hidden, g0, be0, h0, N);

    // ---- layer 1 ----
    k_gemm_f32_wmma<DOUTC, HIDC><<<dim3(gridM, DOUTC / 32), 32, 0, stream>>>(h0, Wl1, bl1, xl1, N);
    k_gemm_f32_wmma<DOUTC, HIDC><<<dim3(gridM, DOUTC / 32), 32, 0, stream>>>(h0, Wr1, br1, xr1, N);

    k_zero<<<((N * DOUTC) + TPB - 1) / TPB, TPB, 0, stream>>>(agg1, N * DOUTC);
    k_zero<<<((N * 2) + TPB - 1) / TPB, TPB, 0, stream>>>((float*)segm, N * 2);
    k_zero<<<((N * 2) + TPB - 1) / TPB, TPB, 0, stream>>>(segs, N * 2);

    k_edge_logits32<<<gridW, TPB, 0, stream>>>(xl1, xr1, srcI, dstI, eattr, We1, att1,
                                               ea_mean, logits, E, N);
    k_seg_max<<<gridE, TPB, 0, stream>>>(logits, dstI, segm, E, N);
    k_seg_expsum<<<gridE, TPB, 0, stream>>>(logits, dstI, segm, segs, E, N);
    k_agg32<<<gridW, TPB, 0, stream>>>(xl1, srcI, dstI, logits, segs, agg1, E, N);

    // out = relu(LN(agg1 + bias1))
    k_final<<<gridN, TPB, 0, stream>>>(agg1, bias1, g1, be1, (float*)d_out, N);
}